// CustomTransformerDecoder_15006615734120
// MI455X (gfx1250) — compile-verified
//
#include <hip/hip_runtime.h>

// ---------------------------------------------------------------------------
// Multi-scale deformable attention decoder layer for gfx1250 (MI455X).
// All dense projections run on V_WMMA_F32_16X16X4_F32 (exact f32 path).
// ---------------------------------------------------------------------------

typedef float v2f __attribute__((ext_vector_type(2)));
typedef float v8f __attribute__((ext_vector_type(8)));

#define NQ    900
#define BSZ   8
#define CDIM  256
#define NHEAD 8
#define NLVL  4
#define NPT   4
#define HDIM  32
#define NVAL  21760   // 128*128 + 64*64 + 32*32 + 16*16

__constant__ __device__ const int   kStart[NLVL] = {0, 16384, 20480, 21504};
__constant__ __device__ const float kHW[NLVL]    = {128.f, 64.f, 32.f, 16.f};

// ---------------------------------------------------------------------------
// Generic WMMA GEMM:  Cout[r, j] = sum_k A[r,k] * W[j,k] + bias[j] (+ resid)
//   A:    rows x 256   (row-major, contiguous)
//   W:    nOut x 256   (row-major; this is "@ W.T" in the reference)
//   nOut: multiple of 64; rows: multiple of 16.
// Each wave computes a 16(M) x 64(N) strip: 4 f32 accumulators, K-loop of 64
// V_WMMA_F32_16X16X4_F32 steps, A-fragment reused across the 4 N-tiles.
// Wave-uniform guard only -> EXEC all-ones at every WMMA (ISA requirement).
// ---------------------------------------------------------------------------
__global__ __launch_bounds__(256)
void gemm_wmma_f32(const float* __restrict__ A, const float* __restrict__ W,
                   const float* __restrict__ bias, const float* __restrict__ resid,
                   float* __restrict__ Cout, int rows, int nOut)
{
    const int lane = threadIdx.x & 31;
    const int wave = (blockIdx.x * blockDim.x + threadIdx.x) >> 5;
    const int tN4  = nOut >> 6;            // 64-wide N strips
    const int tM   = rows >> 4;
    if (wave >= tM * tN4) return;          // drops whole waves only

    const int tn   = wave % tN4;
    const int tm   = wave / tN4;
    const int half = lane >> 4;            // selects K=0,1 vs K=2,3
    const int lm   = lane & 15;            // M (for A) / N (for B,C,D)

    const float* pa  = A + (size_t)(tm * 16 + lm) * CDIM + half * 2;
    const float* pb0 = W + (size_t)(tn * 64 +  0 + lm) * CDIM + half * 2;
    const float* pb1 = pb0 + 16 * CDIM;
    const float* pb2 = pb0 + 32 * CDIM;
    const float* pb3 = pb0 + 48 * CDIM;

    v8f acc0 = {}, acc1 = {}, acc2 = {}, acc3 = {};
#pragma unroll 8
    for (int kk = 0; kk < CDIM / 4; ++kk) {
        v2f a  = *(const v2f*)(pa  + kk * 4);
        v2f b0 = *(const v2f*)(pb0 + kk * 4);
        v2f b1 = *(const v2f*)(pb1 + kk * 4);
        v2f b2 = *(const v2f*)(pb2 + kk * 4);
        v2f b3 = *(const v2f*)(pb3 + kk * 4);
        acc0 = __builtin_amdgcn_wmma_f32_16x16x4_f32(false, a, false, b0, (short)0, acc0, false, false);
        acc1 = __builtin_amdgcn_wmma_f32_16x16x4_f32(false, a, false, b1, (short)0, acc1, false, false);
        acc2 = __builtin_amdgcn_wmma_f32_16x16x4_f32(false, a, false, b2, (short)0, acc2, false, false);
        acc3 = __builtin_amdgcn_wmma_f32_16x16x4_f32(false, a, false, b3, (short)0, acc3, false, false);
    }

    // C/D layout: vgpr i, lanes 0-15 -> (M=i, N=lm); lanes 16-31 -> (M=8+i, N=lm)
    v8f accs[4] = {acc0, acc1, acc2, acc3};
#pragma unroll
    for (int t = 0; t < 4; ++t) {
        const int n  = tn * 64 + t * 16 + lm;
        const float bv = bias[n];
#pragma unroll
        for (int i = 0; i < 8; ++i) {
            const int m = tm * 16 + half * 8 + i;
            float v = accs[t][i] + bv;
            if (resid) v += resid[(size_t)m * CDIM + n];   // only used when nOut==256
            Cout[(size_t)m * nOut + n] = v;
        }
    }
}

// ---------------------------------------------------------------------------
// Softmax over NL*NP = 16 contiguous logits per (row, head).
// ---------------------------------------------------------------------------
__global__ __launch_bounds__(256)
void softmax16_kernel(float* __restrict__ attn, int total)
{
    const int t = blockIdx.x * blockDim.x + threadIdx.x;
    if (t >= total) return;
    float* p = attn + (size_t)t * 16;
    float v[16], m = -3.4e38f;
#pragma unroll
    for (int i = 0; i < 16; ++i) { v[i] = p[i]; m = fmaxf(m, v[i]); }
    float s = 0.f;
#pragma unroll
    for (int i = 0; i < 16; ++i) { v[i] = __expf(v[i] - m); s += v[i]; }
    const float inv = 1.f / s;
#pragma unroll
    for (int i = 0; i < 16; ++i) p[i] = v[i] * inv;
}

// ---------------------------------------------------------------------------
// Bilinear sampling + attention-weighted reduce.
// One wave per (query-row r = iq*BSZ + b, head h); lane = channel d (HDIM=32).
// valp keeps the (nv, bs, C) layout so each corner gather is one coalesced
// 128B read: valp[((start+idx)*BSZ + b)*CDIM + h*32 + d].
// ---------------------------------------------------------------------------
__global__ __launch_bounds__(256)
void msdeform_sample_kernel(const float* __restrict__ valp,
                            const float* __restrict__ off,
                            const float* __restrict__ attn,
                            const float* __restrict__ refp,
                            float* __restrict__ outa)
{
    const int lane = threadIdx.x & 31;
    const int wave = (blockIdx.x * blockDim.x + threadIdx.x) >> 5;
    if (wave >= NQ * BSZ * NHEAD) return;

    const int h  = wave & (NHEAD - 1);
    const int r  = wave >> 3;            // r = iq*BSZ + b
    const int b  = r & (BSZ - 1);
    const int iq = r >> 3;

    const float* offr  = off  + (size_t)r * CDIM + h * (NLVL * NPT * 2);
    const float* attnr = attn + (size_t)r * (NHEAD * NLVL * NPT) + h * (NLVL * NPT);

    float acc = 0.f;
#pragma unroll
    for (int l = 0; l < NLVL; ++l) {
        const float HW   = kHW[l];           // square levels: H == W
        const int   Wi   = (int)HW;
        const int   base = kStart[l];
        const float rx = refp[(((size_t)b * NQ + iq) * NLVL + l) * 2 + 0];
        const float ry = refp[(((size_t)b * NQ + iq) * NLVL + l) * 2 + 1];
#pragma unroll
        for (int p = 0; p < NPT; ++p) {
            const float ox = offr[(l * NPT + p) * 2 + 0];
            const float oy = offr[(l * NPT + p) * 2 + 1];
            const float wa = attnr[l * NPT + p];
            // loc = ref + off/norm ; x = loc*W - 0.5 (match reference assoc.)
            const float x = (rx + ox / HW) * HW - 0.5f;
            const float y = (ry + oy / HW) * HW - 0.5f;
            const float x0f = floorf(x), y0f = floorf(y);
            const float tx = x - x0f, ty = y - y0f;
            const int x0 = (int)x0f, y0 = (int)y0f;
#pragma unroll
            for (int c = 0; c < 4; ++c) {
                const int dx = c & 1, dy = c >> 1;
                const int xi = x0 + dx, yi = y0 + dy;
                const float wx = dx ? tx : 1.f - tx;
                const float wy = dy ? ty : 1.f - ty;
                const bool valid = (xi >= 0) & (xi < Wi) & (yi >= 0) & (yi < Wi);
                const int xc = min(max(xi, 0), Wi - 1);
                const int yc = min(max(yi, 0), Wi - 1);
                const float w = valid ? (wx * wy * wa) : 0.f;
                const size_t row = (size_t)(base + yc * Wi + xc) * BSZ + b;
                acc += w * valp[row * CDIM + h * HDIM + lane];
            }
        }
    }
    outa[(size_t)r * CDIM + h * HDIM + lane] = acc;
}

// ---------------------------------------------------------------------------
// Host launch
// ---------------------------------------------------------------------------
static inline int wave_blocks(long long waves) {
    return (int)((waves * 32 + 255) / 256);
}

extern "C" void kernel_launch(void* const* d_in, const int* in_sizes, int n_in,
                              void* d_out, int out_size, void* d_ws, size_t ws_size,
                              hipStream_t stream) {
    const float* query = (const float*)d_in[0];   // (nq, bs, C)
    const float* value = (const float*)d_in[1];   // (nv, bs, C)
    const float* refp  = (const float*)d_in[2];   // (bs, nq, NL, 2)
    // d_in[3] spatial_shapes: constants hardcoded
    const float* W_off  = (const float*)d_in[4];
    const float* b_off  = (const float*)d_in[5];
    const float* W_attn = (const float*)d_in[6];
    const float* b_attn = (const float*)d_in[7];
    const float* W_val  = (const float*)d_in[8];
    const float* b_val  = (const float*)d_in[9];
    const float* W_out  = (const float*)d_in[10];
    const float* b_out  = (const float*)d_in[11];

    const int qrows = NQ * BSZ;           // 7200
    const int vrows = NVAL * BSZ;         // 174080

    // workspace partition (floats)
    float* val_ws  = (float*)d_ws;                       // vrows*256  = 44.56M
    float* off_ws  = val_ws + (size_t)vrows * CDIM;      // qrows*256  = 1.84M
    float* attn_ws = off_ws + (size_t)qrows * CDIM;      // qrows*128  = 0.92M
    float* outa_ws = attn_ws + (size_t)qrows * 128;      // qrows*256  = 1.84M

    // 1) value projection: val = value @ W_val.T + b_val   (keeps (nv,bs,C) layout)
    gemm_wmma_f32<<<wave_blocks((long long)(vrows / 16) * 4), 256, 0, stream>>>(
        value, W_val, b_val, nullptr, val_ws, vrows, CDIM);

    // 2) sampling-offset projection: off = q @ W_off.T + b_off
    gemm_wmma_f32<<<wave_blocks((long long)(qrows / 16) * 4), 256, 0, stream>>>(
        query, W_off, b_off, nullptr, off_ws, qrows, CDIM);

    // 3) attention-logit projection: attn = q @ W_attn.T + b_attn  (nOut=128)
    gemm_wmma_f32<<<wave_blocks((long long)(qrows / 16) * 2), 256, 0, stream>>>(
        query, W_attn, b_attn, nullptr, attn_ws, qrows, 128);

    // 4) softmax over NL*NP per (row, head)
    {
        const int total = qrows * NHEAD;                 // 57600
        softmax16_kernel<<<(total + 255) / 256, 256, 0, stream>>>(attn_ws, total);
    }

    // 5) bilinear sample + weighted reduce -> (nq*bs, C)
    msdeform_sample_kernel<<<wave_blocks((long long)qrows * NHEAD), 256, 0, stream>>>(
        val_ws, off_ws, attn_ws, refp, outa_ws);

    // 6) output projection + residual: out = outa @ W_out.T + b_out + query
    gemm_wmma_f32<<<wave_blocks((long long)(qrows / 16) * 4), 256, 0, stream>>>(
        outa_ws, W_out, b_out, query, (float*)d_out, qrows, CDIM);
}